// DGRPNModulator_10703058501740
// MI455X (gfx1250) — compile-verified
//
#include <hip/hip_runtime.h>
#include <hip/hip_bf16.h>
#include <math.h>

// ---------------- problem constants (from reference shapes) ----------------
#define B   8
#define C   256
#define H   128
#define W   128
#define HW  (H * W)           // 16384
#define KTOP 50
#define NPART 64              // partial-max blocks for global attn max

typedef float v4f __attribute__((ext_vector_type(4)));
typedef float v2f __attribute__((ext_vector_type(2)));
typedef float v8f __attribute__((ext_vector_type(8)));
typedef int   v4i __attribute__((vector_size(16)));   // matches async builtin param type

#define GAS __attribute__((address_space(1)))
#define LAS __attribute__((address_space(3)))

#if defined(__HIP_DEVICE_COMPILE__) && \
    __has_builtin(__builtin_amdgcn_global_load_async_to_lds_b128) && \
    __has_builtin(__builtin_amdgcn_s_wait_asynccnt)
#define HAVE_ASYNC 1
#else
#define HAVE_ASYNC 0
#endif

#if defined(__HIP_DEVICE_COMPILE__) && \
    __has_builtin(__builtin_amdgcn_wmma_f32_16x16x4_f32)
#define HAVE_WMMA 1
#else
#define HAVE_WMMA 0
#endif

// ---------------------------------------------------------------------------
// Kernel 1: global max of relu(attn) over the whole (B,H,W) tensor.
// ---------------------------------------------------------------------------
__global__ __launch_bounds__(256) void k_attn_max(const float* __restrict__ attn,
                                                  float* __restrict__ partials, int n) {
  __shared__ float red[256];
  float m = 0.0f;  // relu folds in: identity for max after max(x,0)
  for (int i = blockIdx.x * 256 + threadIdx.x; i < n; i += gridDim.x * 256)
    m = fmaxf(m, attn[i]);
  red[threadIdx.x] = m;
  __syncthreads();
  for (int s = 128; s > 0; s >>= 1) {
    if ((int)threadIdx.x < s) red[threadIdx.x] = fmaxf(red[threadIdx.x], red[threadIdx.x + s]);
    __syncthreads();
  }
  if (threadIdx.x == 0) partials[blockIdx.x] = red[0];
}

// ---------------------------------------------------------------------------
// Kernel 2: one workgroup per batch image.
//  - async-stage the 64KB attention map into LDS (ASYNCcnt path)
//  - normalize + threshold, 7x7 NMS, iterative top-50, 9x9 variance
//  - emit peak record {cx, cy, 1/(b*sw^2), 1/(b*sh^2)} to ws
// ---------------------------------------------------------------------------
__global__ __launch_bounds__(256) void k_peaks(const float* __restrict__ attn,
                                               const float* __restrict__ beta_p,
                                               const float* __restrict__ partials,
                                               float* __restrict__ peaks) {
  __shared__ float sM[HW];   // M_th (64 KB)
  __shared__ float sS[HW];   // peak scores (64 KB)
  __shared__ float rV[256];
  __shared__ int   rI[256];
  __shared__ float rX[96];
  __shared__ float rY[96];
  __shared__ float bcast;

  const int b = blockIdx.x, tid = threadIdx.x;
  const float* src = attn + (size_t)b * HW;

  // ---- stage attention map into LDS ----
#if HAVE_ASYNC
  #pragma unroll
  for (int r = 0; r < HW / (256 * 4); ++r) {
    int off = (r * 256 + tid) * 4;
    __builtin_amdgcn_global_load_async_to_lds_b128((GAS v4i*)(src + off),
                                                   (LAS v4i*)(&sM[off]), 0, 0);
  }
  __builtin_amdgcn_s_wait_asynccnt(0);
#else
  for (int r = 0; r < HW / (256 * 4); ++r) {
    int off = (r * 256 + tid) * 4;
    *(v4f*)&sM[off] = *(const v4f*)(src + off);
  }
#endif
  __syncthreads();

  // ---- global max from partials ----
  if (tid < NPART) rV[tid] = partials[tid];
  __syncthreads();
  if (tid == 0) {
    float m = rV[0];
    for (int i = 1; i < NPART; ++i) m = fmaxf(m, rV[i]);
    bcast = m;
  }
  __syncthreads();
  const float mx = bcast;
  const float norm = (mx > 0.0f) ? 1.0f / (mx + 1e-8f) : 1.0f;

  // ---- normalize + threshold in place ----
  for (int p = tid; p < HW; p += 256) {
    float v = fmaxf(sM[p], 0.0f) * norm;
    sM[p] = (v > 0.5f) ? v : 0.0f;
  }
  __syncthreads();

  // ---- 7x7 peak NMS ----
  for (int p = tid; p < HW; p += 256) {
    float v = sM[p], sc = 0.0f;
    if (v > 0.0f) {
      int y = p >> 7, x = p & (W - 1);
      bool pk = true;
      for (int dy = -3; dy <= 3 && pk; ++dy) {
        int yy = y + dy;
        if ((unsigned)yy >= H) continue;
        for (int dx = -3; dx <= 3; ++dx) {
          int xx = x + dx;
          if ((unsigned)xx >= W) continue;
          if (sM[(yy << 7) + xx] > v) { pk = false; break; }
        }
      }
      if (pk) sc = v;
    }
    sS[p] = sc;
  }
  __syncthreads();

  const float bval = fabsf(beta_p[0]) + 1e-6f;

  // ---- iterative top-K with stable (lowest-index) tie-break ----
  for (int k = 0; k < KTOP; ++k) {
    float bv = -2.0f; int bi = 0;
    for (int p = tid; p < HW; p += 256) {
      float v = sS[p];
      if (v > bv || (v == bv && p < bi)) { bv = v; bi = p; }
    }
    rV[tid] = bv; rI[tid] = bi;
    __syncthreads();
    for (int s = 128; s > 0; s >>= 1) {
      if (tid < s) {
        float v = rV[tid + s]; int i2 = rI[tid + s];
        if (v > rV[tid] || (v == rV[tid] && i2 < rI[tid])) { rV[tid] = v; rI[tid] = i2; }
      }
      __syncthreads();
    }
    const int idx = rI[0];
    const int cy = idx >> 7, cx = idx & (W - 1);

    if (tid < 81) {
      int dy = tid / 9 - 4, dx = tid % 9 - 4;
      int yy = cy + dy, xx = cx + dx;
      float wv = ((unsigned)yy < H && (unsigned)xx < W) ? sM[(yy << 7) + xx] : 0.0f;
      rX[tid] = wv * (float)(dx * dx);
      rY[tid] = wv * (float)(dy * dy);
    }
    __syncthreads();
    if (tid == 0) {
      float ssx = 0.0f, ssy = 0.0f;
      for (int i = 0; i < 81; ++i) { ssx += rX[i]; ssy += rY[i]; }
      float sw = fmaxf(sqrtf(fmaxf(ssx, 1e-8f)), 3.0f);
      float sh = fmaxf(sqrtf(fmaxf(ssy, 1e-8f)), 3.0f);
      float* rec = peaks + ((size_t)b * KTOP + k) * 4;
      rec[0] = (float)cx;
      rec[1] = (float)cy;
      rec[2] = 1.0f / (bval * sw * sw);
      rec[3] = 1.0f / (bval * sh * sh);
      sS[idx] = -1.0f;
    }
    __syncthreads();
  }
}

// ---------------------------------------------------------------------------
// Kernel 3 (WMMA): scale(b,h,w) = 1 + gamma * exp(-min_k q_k)
//
// Exact rank-8 factorization: q(p,k) = sum_j a_j(p) * b_j(k) with
//   a(p) = [x^2, x, 1, y^2 | y, 0, 0, 0]
//   b(k) = [invW, -2*invW*cx, invW*cx^2+invH*cy^2, invH | -2*invH*cy, 0, 0, 0]
// => Q = A(16xP tiles) x B(8x64) via two chained V_WMMA_F32_16X16X4_F32
// (K=4 each, accumulate through C). Peaks 50..63 padded with
// b = [0,0,1e30,0,...] so their q is 1e30 and never wins the min.
// D layout (ISA 7.12.2): VGPR r -> pixel r (lanes 0-15) / pixel r+8 (16-31),
// peak = lane%16 -> min across peak groups elementwise, then 4-step
// __shfl_xor min inside each 16-lane half, then one exp per pixel.
// grid = B * 128 blocks, 8 waves/block, 1 wave = one 16-pixel tile.
// ---------------------------------------------------------------------------
__global__ __launch_bounds__(256) void k_scale(const float* __restrict__ peaks,
                                               const float* __restrict__ gamma_p,
                                               float* __restrict__ scale) {
  const int tid   = threadIdx.x;
  const int lane  = tid & 31;
  const int wave  = tid >> 5;
  const int b     = blockIdx.x >> 7;
  const int tile  = ((blockIdx.x & 127) << 3) + wave;   // 0..1023
  const int pbase = tile << 4;                          // 16 pixels per tile
  const int m     = lane & 15;
  const int khalf = lane >> 4;

  const int pix = pbase + m;
  const float fx = (float)(pix & (W - 1));
  const float fy = (float)(pix >> 7);

#if HAVE_WMMA
  // A-matrix (16x4 f32, ISA layout): lanes 0-15 hold K0/K1, lanes 16-31 K2/K3
  v2f a1, a2;
  if (khalf == 0) { a1.x = fx * fx; a1.y = fx;      a2.x = fy;  a2.y = 0.0f; }
  else            { a1.x = 1.0f;    a1.y = fy * fy; a2.x = 0.0f; a2.y = 0.0f; }

  v8f dmin;
  #pragma unroll
  for (int g = 0; g < 4; ++g) {
    const int pid = (g << 4) + m;
    float cx = 0.0f, cy = 0.0f, iw = 0.0f, ih = 0.0f, c2 = 1e30f;
    if (pid < KTOP) {
      const float* rec = peaks + ((size_t)b * KTOP + pid) * 4;
      cx = rec[0]; cy = rec[1]; iw = rec[2]; ih = rec[3];
      c2 = iw * cx * cx + ih * cy * cy;
    }
    // B-matrix (4x16 f32): lanes 0-15 hold rows K0/K1, lanes 16-31 rows K2/K3
    v2f b1, b2;
    if (khalf == 0) { b1.x = iw; b1.y = -2.0f * iw * cx; b2.x = -2.0f * ih * cy; b2.y = 0.0f; }
    else            { b1.x = c2; b1.y = ih;              b2.x = 0.0f;            b2.y = 0.0f; }

    v8f c = {};
    c = __builtin_amdgcn_wmma_f32_16x16x4_f32(false, a1, false, b1, (short)0, c, false, false);
    c = __builtin_amdgcn_wmma_f32_16x16x4_f32(false, a2, false, b2, (short)0, c, false, false);
    if (g == 0) { dmin = c; }
    else {
      #pragma unroll
      for (int r = 0; r < 8; ++r) dmin[r] = fminf(dmin[r], c[r]);
    }
  }
  // min over the 16 peaks held across each 16-lane half
  #pragma unroll
  for (int r = 0; r < 8; ++r) {
    float v = dmin[r];
    #pragma unroll
    for (int s = 1; s < 16; s <<= 1) v = fminf(v, __shfl_xor(v, s, 32));
    dmin[r] = v;
  }
  const float g0 = gamma_p[0];
  float* srow = scale + (size_t)b * HW;
  #pragma unroll
  for (int r = 0; r < 8; ++r) {
    float sv = 1.0f + g0 * __expf(-dmin[r]);
    if (lane == 0)  srow[pbase + r]     = sv;
    if (lane == 16) srow[pbase + r + 8] = sv;
  }
#else
  // scalar fallback: each lane = one pixel, min over all 50 peaks
  float mq = 3.4e38f;
  for (int k = 0; k < KTOP; ++k) {
    const float* rec = peaks + ((size_t)b * KTOP + k) * 4;
    float dx = fx - rec[0], dy = fy - rec[1];
    mq = fminf(mq, dx * dx * rec[2] + dy * dy * rec[3]);
  }
  // both 16-lane halves compute pixel (tile, m) and store the same value
  if (khalf == 0) scale[(size_t)b * HW + pix] = 1.0f + gamma_p[0] * __expf(-mq);
  else            scale[(size_t)b * HW + pix] = 1.0f + gamma_p[0] * __expf(-mq);
#endif
}

// ---------------------------------------------------------------------------
// Kernel 4: out = feats * scale (broadcast over C). Register-cached scale4
// reused across 16 channels; non-temporal B128 for the 256 MiB stream so L2
// keeps the hot 512 KB scale field. grid = 2048 blocks.
// ---------------------------------------------------------------------------
__global__ __launch_bounds__(256) void k_mod(const float* __restrict__ feats,
                                             const float* __restrict__ scale,
                                             float* __restrict__ out) {
  const int bi = blockIdx.x;
  const int b = bi >> 8;
  const int r = bi & 255;
  const int c0 = (r >> 4) << 4;
  const int hw = ((r & 15) << 10) + ((int)threadIdx.x << 2);

  const v4f s4 = *(const v4f*)(scale + (size_t)b * HW + hw);
  const float* fbase = feats + ((size_t)(b * C + c0)) * HW + hw;
  float* obase = out + ((size_t)(b * C + c0)) * HW + hw;

  #pragma unroll
  for (int ci = 0; ci < 16; ++ci) {
    v4f f = __builtin_nontemporal_load((const v4f*)(fbase + (size_t)ci * HW));
    v4f o = f * s4;
    __builtin_nontemporal_store(o, (v4f*)(obase + (size_t)ci * HW));
  }
}

// ---------------------------------------------------------------------------
// ws layout (floats): [0,64) partial maxes | [64,1664) peak records
//                     | [4096, 4096+131072) scale field (B*HW)
// ---------------------------------------------------------------------------
extern "C" void kernel_launch(void* const* d_in, const int* in_sizes, int n_in,
                              void* d_out, int out_size, void* d_ws, size_t ws_size,
                              hipStream_t stream) {
  const float* feats = (const float*)d_in[0];
  const float* attn  = (const float*)d_in[1];
  const float* beta  = (const float*)d_in[2];
  const float* gamma = (const float*)d_in[3];
  float* out = (float*)d_out;
  float* wsf = (float*)d_ws;

  float* partials = wsf;          // 64
  float* peaks    = wsf + 64;     // B*KTOP*4 = 1600
  float* scale    = wsf + 4096;   // B*HW = 131072

  k_attn_max<<<NPART, 256, 0, stream>>>(attn, partials, B * HW);
  k_peaks<<<B, 256, 0, stream>>>(attn, beta, partials, peaks);
  k_scale<<<B * 128, 256, 0, stream>>>(peaks, gamma, scale);
  k_mod<<<B * 16 * 16, 256, 0, stream>>>(feats, scale, out);
}